// GCNOnFeatureMap_14276471292100
// MI455X (gfx1250) — compile-verified
//
#include <hip/hip_runtime.h>
#include <hip/hip_bf16.h>

typedef float v2f __attribute__((ext_vector_type(2)));
typedef float v8f __attribute__((ext_vector_type(8)));

#define CH 128
#define TILE_M 128
#define APITCH 132   // A row pitch in words: 132 % 64 == 4 -> conflict-free strided b64 reads
#define WPITCH 288   // W k-pair row pitch in words: 288 % 64 == 32 -> kh halves hit disjoint banks

// ---------------- elementwise / graph-prep kernels ----------------

__global__ void k_zero(float* __restrict__ p, int n) {
    int i = blockIdx.x * blockDim.x + threadIdx.x;
    if (i < n) p[i] = 0.f;
}

__global__ void k_zero4(float4* __restrict__ p, int n4) {
    int i = blockIdx.x * blockDim.x + threadIdx.x;
    if (i < n4) p[i] = make_float4(0.f, 0.f, 0.f, 0.f);
}

__global__ void k_neg4(float4* __restrict__ p, int n4) {
    int i = blockIdx.x * blockDim.x + threadIdx.x;
    if (i < n4) {
        float4 v = p[i];
        p[i] = make_float4(-v.x, -v.y, -v.z, -v.w);
    }
}

__global__ void k_deg_count(const int* __restrict__ dst, float* __restrict__ deg, int e) {
    int i = blockIdx.x * blockDim.x + threadIdx.x;
    if (i < e) atomicAdd(&deg[dst[i]], 1.0f);
}

// deg -> dis, in place
__global__ void k_dis(float* __restrict__ deg, int n) {
    int i = blockIdx.x * blockDim.x + threadIdx.x;
    if (i < n) {
        float d = deg[i];
        deg[i] = (d > 0.f) ? rsqrtf(fmaxf(d, 1.f)) : 0.f;
    }
}

__global__ void k_edge_w(const int* __restrict__ src, const int* __restrict__ dst,
                         const float* __restrict__ dis, float* __restrict__ nw, int e) {
    int i = blockIdx.x * blockDim.x + threadIdx.x;
    if (i < e) nw[i] = -dis[src[i]] * dis[dst[i]];
}

// y[dst] += scale * nw[e] * x[src]; one thread = (edge, 4 channels), float4 gather
__global__ __launch_bounds__(256)
void k_prop_scatter(const float* __restrict__ x, float* __restrict__ y,
                    const int* __restrict__ src, const int* __restrict__ dst,
                    const float* __restrict__ nw, float scale, int e) {
    int i = blockIdx.x * blockDim.x + threadIdx.x;
    int total = e * (CH / 4);
    if (i < total) {
        int eidx = i >> 5;             // /32 quads per edge
        int q    = i & 31;             // which float4 of the row
        float w = nw[eidx] * scale;
        float4 v = ((const float4*)(x + (long long)src[eidx] * CH))[q];
        float* yp = y + (long long)dst[eidx] * CH + q * 4;
        atomicAdd(yp + 0, w * v.x);
        atomicAdd(yp + 1, w * v.y);
        atomicAdd(yp + 2, w * v.z);
        atomicAdd(yp + 3, w * v.w);
    }
}

// ---------------- BatchNorm ----------------

__global__ __launch_bounds__(256)
void k_bn_reduce(const float* __restrict__ x, float* __restrict__ gsum,
                 float* __restrict__ gsq, int n) {
    __shared__ float4 ssum[256];
    __shared__ float4 ssq[256];
    int t  = threadIdx.x;
    int q  = t & 31;                   // float4 column (channels 4q..4q+3)
    int rs = t >> 5;                   // row slice 0..7
    const int ROWS = 128;
    int r0 = blockIdx.x * ROWS;
    int rend = r0 + ROWS; if (rend > n) rend = n;
    float4 s = make_float4(0.f, 0.f, 0.f, 0.f);
    float4 sq2 = make_float4(0.f, 0.f, 0.f, 0.f);
    for (int r = r0 + rs; r < rend; r += 8) {
        float4 v = ((const float4*)(x + (long long)r * CH))[q];
        s.x += v.x; s.y += v.y; s.z += v.z; s.w += v.w;
        sq2.x += v.x * v.x; sq2.y += v.y * v.y; sq2.z += v.z * v.z; sq2.w += v.w * v.w;
    }
    ssum[t] = s; ssq[t] = sq2;
    __syncthreads();
    if (t < 32) {
        float4 S = ssum[t], Q = ssq[t];
        #pragma unroll
        for (int j = 1; j < 8; ++j) {
            float4 a = ssum[t + 32 * j], b = ssq[t + 32 * j];
            S.x += a.x; S.y += a.y; S.z += a.z; S.w += a.w;
            Q.x += b.x; Q.y += b.y; Q.z += b.z; Q.w += b.w;
        }
        atomicAdd(&gsum[t * 4 + 0], S.x); atomicAdd(&gsum[t * 4 + 1], S.y);
        atomicAdd(&gsum[t * 4 + 2], S.z); atomicAdd(&gsum[t * 4 + 3], S.w);
        atomicAdd(&gsq[t * 4 + 0], Q.x);  atomicAdd(&gsq[t * 4 + 1], Q.y);
        atomicAdd(&gsq[t * 4 + 2], Q.z);  atomicAdd(&gsq[t * 4 + 3], Q.w);
    }
}

__global__ void k_bn_final(const float* __restrict__ gsum, const float* __restrict__ gsq,
                           const float* __restrict__ gamma, const float* __restrict__ beta,
                           float* __restrict__ scale, float* __restrict__ shift, float inv_n) {
    int c = threadIdx.x;
    if (c < CH) {
        float mean = gsum[c] * inv_n;
        float var  = gsq[c] * inv_n - mean * mean;
        float sc = gamma[c] * rsqrtf(var + 1e-5f);
        scale[c] = sc;
        shift[c] = beta[c] - mean * sc;
    }
}

__global__ void k_bn_apply_relu(const float4* __restrict__ x, const float* __restrict__ scale,
                                const float* __restrict__ shift, float4* __restrict__ y, int n4) {
    int i = blockIdx.x * blockDim.x + threadIdx.x;
    if (i < n4) {
        int c = (i * 4) & (CH - 1);
        float4 v = x[i];
        float a = v.x * scale[c + 0] + shift[c + 0];
        float b = v.y * scale[c + 1] + shift[c + 1];
        float cc = v.z * scale[c + 2] + shift[c + 2];
        float d = v.w * scale[c + 3] + shift[c + 3];
        y[i] = make_float4(a > 0.f ? a : 0.f, b > 0.f ? b : 0.f,
                           cc > 0.f ? cc : 0.f, d > 0.f ? d : 0.f);
    }
}

// ---------------- WMMA f32 GEMM: out[n x 128] (+)= A[n x 128] @ W[128 x 128] (+bias) ----------------
// 256 threads = 8 wave32; block tile 128(M) x 128(N); each wave owns a 16x128 strip
// (8 x v8f accumulators of V_WMMA_F32_16X16X4_F32). Full A tile and full W staged in
// dynamic LDS with fragment-friendly layouts so every fragment is ONE aligned ds_load_b64:
//   A: row-major, pitch 132 words       -> a-frag = float2 at [row][kk+2*kh]
//   W: k-pair interleaved, pitch 288    -> b-frag = float2 (W[k][n],W[k+1][n]) at [kp][n]
// One barrier per block; 32 k-steps x 8 WMMAs back-to-back.

__global__ __launch_bounds__(256)
void k_gemm128(const float* __restrict__ A, const float* __restrict__ W,
               const float* __restrict__ bias, float* __restrict__ out,
               int n, int accumulate) {
    extern __shared__ float smem[];
    float* As = smem;                          // 128 * 132 words
    float* Ws = smem + TILE_M * APITCH;        // 64  * 288 words

    const int tid  = threadIdx.x;
    const int lane = tid & 31;
    const int wave = tid >> 5;                 // 0..7
    const int m16  = lane & 15;                // M (A frag) / N (B,C,D frags)
    const int kh   = lane >> 4;                // K-pair half / row-half of C
    const int row0 = blockIdx.x * TILE_M;

    // ---- stage A tile (128 x 128) via float4 global loads ----
    for (int i = tid; i < TILE_M * (CH / 4); i += 256) {
        int r = i >> 5, q = i & 31;
        int gr = row0 + r;
        float4 v = (gr < n) ? ((const float4*)(A + (long long)gr * CH))[q]
                            : make_float4(0.f, 0.f, 0.f, 0.f);
        *((float4*)&As[r * APITCH + q * 4]) = v;   // 528B row pitch -> 16B aligned
    }
    // ---- stage W (128 x 128) as k-pair float2s: Ws[kp*WPITCH + n*2] = {W[2kp][n], W[2kp+1][n]} ----
    for (int i = tid; i < 64 * CH; i += 256) {
        int kp = i >> 7, c = i & 127;
        v2f w2;
        w2.x = W[(2 * kp) * CH + c];
        w2.y = W[(2 * kp + 1) * CH + c];
        *((v2f*)&Ws[kp * WPITCH + c * 2]) = w2;    // conflict-free b64 stores (word stride 2)
    }

    v8f acc[8];
    if (accumulate) {
        #pragma unroll
        for (int nt = 0; nt < 8; ++nt)
            #pragma unroll
            for (int j = 0; j < 8; ++j) {
                int r = row0 + wave * 16 + kh * 8 + j;
                acc[nt][j] = (r < n) ? out[(long long)r * CH + nt * 16 + m16] : 0.f;
            }
    } else {
        #pragma unroll
        for (int nt = 0; nt < 8; ++nt) {
            float bv = bias ? bias[nt * 16 + m16] : 0.f;
            #pragma unroll
            for (int j = 0; j < 8; ++j) acc[nt][j] = bv;
        }
    }

    __syncthreads();

    const float* arow = &As[(wave * 16 + m16) * APITCH];
    for (int kk = 0; kk < CH; kk += 4) {
        // A frag: lanes 0-15 hold K={kk,kk+1}, lanes 16-31 K={kk+2,kk+3}; 8B-aligned b64
        v2f a = *((const v2f*)&arow[kk + 2 * kh]);
        const float* wrow = &Ws[((kk >> 1) + kh) * WPITCH + m16 * 2];
        #pragma unroll
        for (int nt = 0; nt < 8; ++nt) {
            v2f b = *((const v2f*)&wrow[nt * 32]);   // immediate-offset b64 loads
            acc[nt] = __builtin_amdgcn_wmma_f32_16x16x4_f32(
                false, a, false, b, (short)0, acc[nt], false, false);
        }
    }

    #pragma unroll
    for (int nt = 0; nt < 8; ++nt)
        #pragma unroll
        for (int j = 0; j < 8; ++j) {
            int r = row0 + wave * 16 + kh * 8 + j;
            if (r < n) out[(long long)r * CH + nt * 16 + m16] = acc[nt][j];
        }
}

// ---------------- host orchestration ----------------

static inline int cdiv(int a, int b) { return (a + b - 1) / b; }

extern "C" void kernel_launch(void* const* d_in, const int* in_sizes, int n_in,
                              void* d_out, int out_size, void* d_ws, size_t ws_size,
                              hipStream_t stream) {
    const float* x      = (const float*)d_in[0];
    const int*   ei     = (const int*)  d_in[1];
    const float* w1     = (const float*)d_in[2];   // [1,128,128]
    const float* b1     = (const float*)d_in[3];
    const float* w2     = (const float*)d_in[4];   // [3,128,128]
    const float* b2     = (const float*)d_in[5];
    const float* w3     = (const float*)d_in[6];   // [5,128,128]
    const float* b3     = (const float*)d_in[7];
    const float* gamma1 = (const float*)d_in[8];
    const float* beta1  = (const float*)d_in[9];
    const float* gamma2 = (const float*)d_in[10];
    const float* beta2  = (const float*)d_in[11];
    float* outp = (float*)d_out;

    const int N = in_sizes[0] / CH;
    const int E = in_sizes[1] / 2;
    const int* srcIdx = ei;        // edge_index[0]
    const int* dstIdx = ei + E;    // edge_index[1]

    const int NF  = N * CH;        // feature-map elements
    const int NF4 = NF / 4;
    const float inv_n = 1.0f / (float)N;

    // workspace layout (floats)
    float* ws   = (float*)d_ws;
    float* deg  = ws;                       // N  (becomes dis in place)
    float* nw   = deg + N;                  // E
    float* buf0 = nw + E;                   // N*128
    float* buf1 = buf0 + (size_t)NF;        // N*128
    float* bufA = buf1 + (size_t)NF;        // N*128 (pre-BN accumulator)
    float* gsum = bufA + (size_t)NF;        // 128
    float* gsq  = gsum + CH;                // 128
    float* bnsc = gsq + CH;                 // 128
    float* bnsh = bnsc + CH;                // 128

    const int B = 256;
    const size_t SMEM = (size_t)(TILE_M * APITCH + 64 * WPITCH) * sizeof(float); // 141,312 B
    const int gN   = cdiv(N, B);
    const int gE   = cdiv(E, B);
    const int gNF4 = cdiv(NF4, B);
    const int gEC4 = cdiv(E * (CH / 4), B);
    const int gGemm = cdiv(N, TILE_M);
    const int gBn   = cdiv(N, 128);

    // ---- edge normalization ----
    k_zero<<<gN, B, 0, stream>>>(deg, N);
    k_deg_count<<<gE, B, 0, stream>>>(dstIdx, deg, E);
    k_dis<<<gN, B, 0, stream>>>(deg, N);
    k_edge_w<<<gE, B, 0, stream>>>(srcIdx, dstIdx, deg, nw, E);

    // ---- layer 1: K=1 ---- bufA = x @ W1 + b1
    k_gemm128<<<gGemm, B, SMEM, stream>>>(x, w1, b1, bufA, N, 0);
    // BN1 + ReLU -> buf0
    k_zero<<<1, B, 0, stream>>>(gsum, 2 * CH);
    k_bn_reduce<<<gBn, B, 0, stream>>>(bufA, gsum, gsq, N);
    k_bn_final<<<1, CH, 0, stream>>>(gsum, gsq, gamma1, beta1, bnsc, bnsh, inv_n);
    k_bn_apply_relu<<<gNF4, B, 0, stream>>>((const float4*)bufA, bnsc, bnsh, (float4*)buf0, NF4);

    // ---- layer 2: K=3 ---- Tx0 = buf0
    k_gemm128<<<gGemm, B, SMEM, stream>>>(buf0, w2, b2, bufA, N, 0);               // + Tx0@W2[0]
    k_zero4<<<gNF4, B, 0, stream>>>((float4*)buf1, NF4);
    k_prop_scatter<<<gEC4, B, 0, stream>>>(buf0, buf1, srcIdx, dstIdx, nw, 1.f, E); // Tx1
    k_gemm128<<<gGemm, B, SMEM, stream>>>(buf1, w2 + CH * CH, nullptr, bufA, N, 1); // + Tx1@W2[1]
    k_neg4<<<gNF4, B, 0, stream>>>((float4*)buf0, NF4);                             // buf0 = -Tx0
    k_prop_scatter<<<gEC4, B, 0, stream>>>(buf1, buf0, srcIdx, dstIdx, nw, 2.f, E); // Tx2 = 2*prop(Tx1)-Tx0
    k_gemm128<<<gGemm, B, SMEM, stream>>>(buf0, w2 + 2 * CH * CH, nullptr, bufA, N, 1);
    // BN2 + ReLU -> buf0 (becomes layer-3 Tx0)
    k_zero<<<1, B, 0, stream>>>(gsum, 2 * CH);
    k_bn_reduce<<<gBn, B, 0, stream>>>(bufA, gsum, gsq, N);
    k_bn_final<<<1, CH, 0, stream>>>(gsum, gsq, gamma2, beta2, bnsc, bnsh, inv_n);
    k_bn_apply_relu<<<gNF4, B, 0, stream>>>((const float4*)bufA, bnsc, bnsh, (float4*)buf0, NF4);

    // ---- layer 3: K=5 ---- accumulate directly into d_out
    k_gemm128<<<gGemm, B, SMEM, stream>>>(buf0, w3, b3, outp, N, 0);               // + Tx0@W3[0]
    k_zero4<<<gNF4, B, 0, stream>>>((float4*)buf1, NF4);
    k_prop_scatter<<<gEC4, B, 0, stream>>>(buf0, buf1, srcIdx, dstIdx, nw, 1.f, E); // Tx1
    k_gemm128<<<gGemm, B, SMEM, stream>>>(buf1, w3 + CH * CH, nullptr, outp, N, 1);
    k_neg4<<<gNF4, B, 0, stream>>>((float4*)buf0, NF4);
    k_prop_scatter<<<gEC4, B, 0, stream>>>(buf1, buf0, srcIdx, dstIdx, nw, 2.f, E); // Tx2
    k_gemm128<<<gGemm, B, SMEM, stream>>>(buf0, w3 + 2 * CH * CH, nullptr, outp, N, 1);
    k_neg4<<<gNF4, B, 0, stream>>>((float4*)buf1, NF4);
    k_prop_scatter<<<gEC4, B, 0, stream>>>(buf0, buf1, srcIdx, dstIdx, nw, 2.f, E); // Tx3
    k_gemm128<<<gGemm, B, SMEM, stream>>>(buf1, w3 + 3 * CH * CH, nullptr, outp, N, 1);
    k_neg4<<<gNF4, B, 0, stream>>>((float4*)buf0, NF4);
    k_prop_scatter<<<gEC4, B, 0, stream>>>(buf1, buf0, srcIdx, dstIdx, nw, 2.f, E); // Tx4
    k_gemm128<<<gGemm, B, SMEM, stream>>>(buf0, w3 + 4 * CH * CH, nullptr, outp, N, 1);
}